// GCNLayer_38723425141161
// MI455X (gfx1250) — compile-verified
//
#include <hip/hip_runtime.h>
#include <hip/hip_bf16.h>

typedef __attribute__((ext_vector_type(2))) float v2f;
typedef __attribute__((ext_vector_type(8))) float v8f;

#define F_IN  64
#define UNITS 64

// ---------------------------------------------------------------------------
// Kernel 1: XW = X @ W  using V_WMMA_F32_16X16X4_F32 (fp32 matrix pipe).
// One wave computes a 16-row x 64-col strip of XW (4 accumulator tiles).
//
// W is staged in LDS *pre-interleaved by K-pairs*:
//   wlds[kp*64 + n] = (W[2kp][n], W[2kp+1][n])  as float2
// so every B-fragment {W[ka][n], W[ka+1][n]} is ONE aligned ds_load_b64 that
// lands directly in an even VGPR pair -> no v_mov marshalling before wmma.
//
// ISA layouts (CDNA5 ISA §7.12.2):
//   A 16x4 f32:  lanes 0-15 -> M=lane, K=0(v0),1(v1); lanes 16-31 -> M=lane-16, K=2,3
//   B 4x16 f32:  lanes 0-15 -> N=lane, K=0(v0),1(v1); lanes 16-31 -> N=lane-16, K=2,3
//   C/D 16x16:   VGPR i -> M=i (lanes 0-15) / M=8+i (lanes 16-31), N = lane%16
// ---------------------------------------------------------------------------
__global__ void __launch_bounds__(128) gcn_gemm_xw(
    const float* __restrict__ X, const float* __restrict__ W,
    float* __restrict__ XW, int nTiles)
{
    __shared__ float2 wlds[(F_IN / 2) * UNITS];   // 32 k-pairs x 64 cols = 16 KB
    for (int i = threadIdx.x; i < (F_IN / 2) * UNITS; i += 128) {
        const int kp = i >> 6;          // k-pair index (k = 2*kp, 2*kp+1)
        const int n  = i & 63;
        wlds[i] = make_float2(W[(2 * kp) * UNITS + n], W[(2 * kp + 1) * UNITS + n]);
    }
    __syncthreads();

    const int lane = threadIdx.x & 31;
    const int wave = threadIdx.x >> 5;
    const int tile = blockIdx.x * 4 + wave;   // wave-uniform
    if (tile >= nTiles) return;               // EXEC stays all-ones per wave

    const int l16 = lane & 15;
    const int hi  = lane >> 4;                // 0: K pair {0,1}   1: K pair {2,3}
    const int row = tile * 16 + l16;

    const float* __restrict__ xrow = X + (size_t)row * F_IN;

    v8f acc[4];
    #pragma unroll
    for (int nt = 0; nt < 4; ++nt) acc[nt] = (v8f){0.f,0.f,0.f,0.f,0.f,0.f,0.f,0.f};

    #pragma unroll
    for (int ks = 0; ks < F_IN / 4; ++ks) {
        const int ka = ks * 4 + hi * 2;       // this lane's first K
        const int kp = ks * 2 + hi;           // k-pair index in LDS image
        v2f a;
        a.x = xrow[ka];                       // one global_load_b64
        a.y = xrow[ka + 1];

        #pragma unroll
        for (int nt = 0; nt < 4; ++nt) {
            const float2 bb = wlds[kp * UNITS + nt * 16 + l16];  // one ds_load_b64
            v2f b;
            b.x = bb.x;
            b.y = bb.y;
            // 8 args: (neg_a, A, neg_b, B, c_mod, C, reuse_a, reuse_b)
            acc[nt] = __builtin_amdgcn_wmma_f32_16x16x4_f32(
                false, a, false, b, (short)0, acc[nt], false, false);
        }
    }

    float* __restrict__ orow = XW + (size_t)tile * 16 * UNITS;
    #pragma unroll
    for (int i = 0; i < 8; ++i) {
        const int m = i + hi * 8;
        #pragma unroll
        for (int nt = 0; nt < 4; ++nt)
            orow[m * UNITS + nt * 16 + l16] = acc[nt][i];
    }
}

// ---------------------------------------------------------------------------
// Kernel 2: out[n, u] = b[u]  (bias pre-init so the scatter is pure adds)
// ---------------------------------------------------------------------------
__global__ void __launch_bounds__(256) gcn_init_bias(
    const float* __restrict__ b, float* __restrict__ out, int total)
{
    int i = blockIdx.x * 256 + threadIdx.x;
    if (i < total) out[i] = b[i & (UNITS - 1)];
}

// ---------------------------------------------------------------------------
// Kernel 3: for each edge e:  out[row[e], :] += weight[e] * XW[col[e], :]
// 16 lanes per edge, float4 per lane, hardware f32 atomics (resolve in L2:
// the 25.6 MB output + most of XW fit in the 192 MB device L2).
// ---------------------------------------------------------------------------
__global__ void __launch_bounds__(256) gcn_scatter_edges(
    const float* __restrict__ xw, const float* __restrict__ ew,
    const int* __restrict__ erow, const int* __restrict__ ecol,
    float* __restrict__ out, int nEdges)
{
    const int g = threadIdx.x >> 4;           // edge slot within block (0..15)
    const int l = threadIdx.x & 15;           // lane within edge
    const int e = blockIdx.x * 16 + g;
    if (e >= nEdges) return;

    const int   col = ecol[e];
    const int   row = erow[e];
    const float w   = ew[e];

    const float4 v = ((const float4*)(xw + (size_t)col * UNITS))[l];
    float* dst = out + (size_t)row * UNITS + l * 4;
    unsafeAtomicAdd(dst + 0, v.x * w);
    unsafeAtomicAdd(dst + 1, v.y * w);
    unsafeAtomicAdd(dst + 2, v.z * w);
    unsafeAtomicAdd(dst + 3, v.w * w);
}

// ---------------------------------------------------------------------------
extern "C" void kernel_launch(void* const* d_in, const int* in_sizes, int n_in,
                              void* d_out, int out_size, void* d_ws, size_t ws_size,
                              hipStream_t stream)
{
    const float* x    = (const float*)d_in[0];   // [N, 64]
    const float* w    = (const float*)d_in[1];   // [64, 64]
    const float* bias = (const float*)d_in[2];   // [64]
    const float* ew   = (const float*)d_in[3];   // [E]
    const int*   erow = (const int*)  d_in[4];   // [E]
    const int*   ecol = (const int*)  d_in[5];   // [E]
    float*       out  = (float*)d_out;           // [N, 64]
    float*       xw   = (float*)d_ws;            // [N, 64] scratch (25.6 MB)

    const int N = in_sizes[0] / F_IN;            // 100000 (multiple of 16)
    const int E = in_sizes[3];                   // 800000

    const int nTiles = N / 16;                                    // 6250
    dim3 gemmGrid((nTiles + 3) / 4);
    gcn_gemm_xw<<<gemmGrid, 128, 0, stream>>>(x, w, xw, nTiles);

    const int total = N * UNITS;
    gcn_init_bias<<<(total + 255) / 256, 256, 0, stream>>>(bias, out, total);

    gcn_scatter_edges<<<(E + 15) / 16, 256, 0, stream>>>(xw, ew, erow, ecol, out, E);
}